// IterNormRotation_24575802868415
// MI455X (gfx1250) — compile-verified
//
#include <hip/hip_runtime.h>
#include <math.h>

// Problem constants (B, C, H, W) = (32, 256, 56, 56), g = 1, nc = 256
#define BATCH 32
#define CHN   256
#define HW    3136               // 56*56
#define MTOT  (BATCH * HW)       // 100352
#define EPSV  1e-5f

typedef __attribute__((ext_vector_type(2))) float v2f;
typedef __attribute__((ext_vector_type(4))) float v4f;
typedef __attribute__((ext_vector_type(8))) float v8f;

// D = A(16x4) * B(4x16) + C, f32 WMMA (wave32).
// Fragment mapping (from ISA layout tables):
//   A: lane l, vgpr v -> A[row = l&15][k = 2*(l>>4) + v]
//   B: lane l, vgpr v -> B[k = 2*(l>>4) + v][col = l&15]
//   C/D: lane l, vgpr v -> C[row = v + 8*(l>>4)][col = l&15]
__device__ inline v8f wmma_f32_k4(v2f a, v2f b, v8f c) {
  return __builtin_amdgcn_wmma_f32_16x16x4_f32(
      /*neg_a=*/false, a, /*neg_b=*/false, b,
      /*c_mod=*/(short)0, c, /*reuse_a=*/false, /*reuse_b=*/false);
}

// ---------------------------------------------------------------------------
// 1) Per-channel mean over m = B*H*W (pure bandwidth pass; b128 loads)
// ---------------------------------------------------------------------------
__global__ __launch_bounds__(256) void mean_kernel(const float* __restrict__ X,
                                                   float* __restrict__ mean) {
  const int c = blockIdx.x;
  const int t = threadIdx.x;
  float s = 0.0f;
  for (int b = 0; b < BATCH; ++b) {
    const v4f* p = (const v4f*)(X + ((size_t)b * CHN + c) * HW);
    for (int i = t; i < HW / 4; i += 256) {
      v4f q = p[i];
      s += q.x + q.y + q.z + q.w;
    }
  }
  __shared__ float red[256];
  red[t] = s;
  __syncthreads();
  for (int k = 128; k > 0; k >>= 1) {
    if (t < k) red[t] += red[t + k];
    __syncthreads();
  }
  if (t == 0) mean[c] = red[0] * (1.0f / (float)MTOT);
}

// ---------------------------------------------------------------------------
// 2) Gram matrix G = x @ x^T (symmetric: only 136 upper-triangular tiles).
//    One wave per (tile pair, batch-slice); atomicAdd partial tiles + mirror.
// ---------------------------------------------------------------------------
__global__ __launch_bounds__(32) void gram_kernel(const float* __restrict__ X,
                                                  float* __restrict__ G) {
  // Map pair index p (0..135) -> (ti, tj) with ti <= tj (scalar, <=16 iters)
  int p = blockIdx.x;
  int ti = 0;
  while (p >= 16 - ti) { p -= 16 - ti; ++ti; }
  const int tj = ti + p;
  const int b = blockIdx.y;

  const int lane = threadIdx.x;
  const int r = lane & 15;      // A row / B col
  const int h = lane >> 4;      // k-half selector
  const float* xb = X + (size_t)b * CHN * HW;
  const float* arow = xb + (size_t)(ti * 16 + r) * HW + 2 * h;  // x[i-row][k..]
  const float* brow = xb + (size_t)(tj * 16 + r) * HW + 2 * h;  // x[j-col][k..]
  v8f acc = {};
#pragma unroll 4
  for (int k = 0; k < HW; k += 4) {
    v2f a = *(const v2f*)(arow + k);
    v2f bb = *(const v2f*)(brow + k);
    acc = wmma_f32_k4(a, bb, acc);
  }
#pragma unroll
  for (int v = 0; v < 8; ++v) {
    const int i = ti * 16 + v + 8 * h;
    const int j = tj * 16 + r;
    atomicAdd(&G[i * CHN + j], acc[v]);
    if (ti != tj) atomicAdd(&G[j * CHN + i], acc[v]);  // mirror (symmetry)
  }
}

// ---------------------------------------------------------------------------
// 3) Sigma_N = rTr * (eps*I + G/m - mean*mean^T); P = I; store rTr, sqrt(rTr)
// ---------------------------------------------------------------------------
__global__ __launch_bounds__(256) void finalize_kernel(const float* __restrict__ G,
                                                       const float* __restrict__ mean,
                                                       float* __restrict__ SigmaN,
                                                       float* __restrict__ P,
                                                       float* __restrict__ scal) {
  const int t = threadIdx.x;
  const float invm = 1.0f / (float)MTOT;
  const float mt = mean[t];
  __shared__ float red[256];
  red[t] = EPSV + G[t * CHN + t] * invm - mt * mt;  // diag(Sigma)
  __syncthreads();
  for (int k = 128; k > 0; k >>= 1) {
    if (t < k) red[t] += red[t + k];
    __syncthreads();
  }
  __shared__ float srTr;
  if (t == 0) {
    const float r = 1.0f / red[0];
    scal[0] = r;
    scal[1] = sqrtf(r);
    srTr = r;
  }
  __syncthreads();
  const float rTr = srTr;
  for (int i = 0; i < CHN; ++i) {
    float v = G[i * CHN + t] * invm - mean[i] * mt;
    if (i == t) v += EPSV;
    SigmaN[i * CHN + t] = v * rTr;
    P[i * CHN + t] = (i == t) ? 1.0f : 0.0f;
  }
}

// ---------------------------------------------------------------------------
// 4) 256x256x256 matmul: D = alpha*(A@B)*(scale?*scale:1) + beta*E
//    One wave per 16x16 output tile (grid = 256).
// ---------------------------------------------------------------------------
__global__ __launch_bounds__(32) void matmul256_kernel(const float* __restrict__ A,
                                                       const float* __restrict__ B,
                                                       const float* __restrict__ E,
                                                       float* __restrict__ D,
                                                       float alpha, float beta,
                                                       const float* __restrict__ scale) {
  const int tile = blockIdx.x;
  const int ti = tile >> 4, tj = tile & 15;
  const int lane = threadIdx.x;
  const int r = lane & 15, h = lane >> 4;
  const float* arow = A + (ti * 16 + r) * CHN + 2 * h;
  const float* bcol = B + tj * 16 + r;
  v8f acc = {};
#pragma unroll 4
  for (int k = 0; k < CHN; k += 4) {
    v2f a = *(const v2f*)(arow + k);
    v2f bb;
    bb.x = bcol[(k + 2 * h) * CHN];
    bb.y = bcol[(k + 2 * h + 1) * CHN];
    acc = wmma_f32_k4(a, bb, acc);
  }
  const float al = alpha * (scale ? scale[0] : 1.0f);
#pragma unroll
  for (int v = 0; v < 8; ++v) {
    const int i = ti * 16 + v + 8 * h;
    const int j = tj * 16 + r;
    float e = (beta != 0.0f) ? E[i * CHN + j] : 0.0f;
    D[i * CHN + j] = al * acc[v] + beta * e;
  }
}

// ---------------------------------------------------------------------------
// 5) Fold weight/bias/mean: Mp[d][c] = w[d]*M0[d][c];
//    betav[d] = bias[d] - w[d] * dot(M0[d,:], mean)
// ---------------------------------------------------------------------------
__global__ __launch_bounds__(256) void fold_kernel(const float* __restrict__ M0,
                                                   const float* __restrict__ mean,
                                                   const float* __restrict__ weight,
                                                   const float* __restrict__ bias,
                                                   float* __restrict__ Mp,
                                                   float* __restrict__ betav) {
  const int d = threadIdx.x;
  const float w = weight[d];
  float dot = 0.0f;
  for (int c = 0; c < CHN; ++c) {
    const float m0 = M0[d * CHN + c];
    dot += m0 * mean[c];
    Mp[d * CHN + c] = m0 * w;
  }
  betav[d] = bias[d] - dot * w;
}

// ---------------------------------------------------------------------------
// 6) Apply: out[b,d,hw] = sum_c Mp[d][c] * X[b,c,hw] + betav[d]
//    One wave per (group of 4 d-tiles, 16-wide hw chunk): each strided B
//    fragment of X is loaded once and fed to 4 WMMAs (4x less X traffic).
// ---------------------------------------------------------------------------
__global__ __launch_bounds__(256) void apply_kernel(const float* __restrict__ X,
                                                    const float* __restrict__ Mp,
                                                    const float* __restrict__ betav,
                                                    float* __restrict__ out) {
  const int wid = blockIdx.x * 8 + (threadIdx.x >> 5);
  const int lane = threadIdx.x & 31;
  const int r = lane & 15, h = lane >> 4;
  const int tg = wid & 3;             // d-tiles 4*tg .. 4*tg+3 (64 channels)
  const int chunk = wid >> 2;         // 0 .. 6271
  const int b = chunk / (HW / 16);    // 196 chunks per batch image
  const int hw0 = (chunk % (HW / 16)) * 16;

  const float* a0 = Mp + ((tg * 4 + 0) * 16 + r) * CHN + 2 * h;
  const float* a1 = a0 + 16 * CHN;
  const float* a2 = a1 + 16 * CHN;
  const float* a3 = a2 + 16 * CHN;
  const float* xb = X + (size_t)b * CHN * HW + hw0 + r;  // column n = r
  v8f acc0 = {}, acc1 = {}, acc2 = {}, acc3 = {};
#pragma unroll 2
  for (int k = 0; k < CHN; k += 4) {
    v2f bb;
    bb.x = xb[(k + 2 * h) * HW];
    bb.y = xb[(k + 2 * h + 1) * HW];
    acc0 = wmma_f32_k4(*(const v2f*)(a0 + k), bb, acc0);
    acc1 = wmma_f32_k4(*(const v2f*)(a1 + k), bb, acc1);
    acc2 = wmma_f32_k4(*(const v2f*)(a2 + k), bb, acc2);
    acc3 = wmma_f32_k4(*(const v2f*)(a3 + k), bb, acc3);
  }
  float* ob = out + (size_t)b * CHN * HW + hw0 + r;
#pragma unroll
  for (int v = 0; v < 8; ++v) {
    const int dl = v + 8 * h;
    int d0 = (tg * 4 + 0) * 16 + dl;
    int d1 = (tg * 4 + 1) * 16 + dl;
    int d2 = (tg * 4 + 2) * 16 + dl;
    int d3 = (tg * 4 + 3) * 16 + dl;
    ob[d0 * HW] = acc0[v] + betav[d0];
    ob[d1 * HW] = acc1[v] + betav[d1];
    ob[d2 * HW] = acc2[v] + betav[d2];
    ob[d3 * HW] = acc3[v] + betav[d3];
  }
}

// ---------------------------------------------------------------------------
// Host-side orchestration (graph-capture safe: only async calls on stream)
// ---------------------------------------------------------------------------
extern "C" void kernel_launch(void* const* d_in, const int* in_sizes, int n_in,
                              void* d_out, int out_size, void* d_ws, size_t ws_size,
                              hipStream_t stream) {
  const float* X      = (const float*)d_in[0];
  const float* rot    = (const float*)d_in[1];   // [1,256,256]
  const float* weight = (const float*)d_in[2];   // [1,256,1,1]
  const float* bias   = (const float*)d_in[3];   // [1,256,1,1]
  float* outp = (float*)d_out;

  float* ws = (float*)d_ws;
  const int NN = CHN * CHN;  // 65536
  float* mean   = ws;              // 256
  float* G      = mean + 256;      // 65536
  float* SigmaN = G + NN;          // 65536
  float* P0     = SigmaN + NN;     // 65536
  float* P1     = P0 + NN;         // 65536
  float* T1     = P1 + NN;         // 65536
  float* T2     = T1 + NN;         // 65536
  float* M0     = T2 + NN;         // 65536
  float* Mp     = M0 + NN;         // 65536
  float* betav  = Mp + NN;         // 256
  float* scal   = betav + 256;     // 2: [rTr, sqrt(rTr)]

  // G accumulated via atomics -> must be zeroed every launch
  hipMemsetAsync(G, 0, (size_t)NN * sizeof(float), stream);

  mean_kernel<<<CHN, 256, 0, stream>>>(X, mean);
  // 136 upper-triangular tile pairs x 32 batch slices, one wave each
  gram_kernel<<<dim3(136, BATCH), 32, 0, stream>>>(X, G);
  finalize_kernel<<<1, 256, 0, stream>>>(G, mean, SigmaN, P0, scal);

  // Newton-Schulz: P <- 1.5 P - 0.5 * P*(P*P)*Sigma_N, T = 5
  float* Pc = P0;
  float* Pn = P1;
  for (int it = 0; it < 5; ++it) {
    matmul256_kernel<<<256, 32, 0, stream>>>(Pc, Pc, nullptr, T1, 1.0f, 0.0f, nullptr);
    matmul256_kernel<<<256, 32, 0, stream>>>(Pc, T1, nullptr, T2, 1.0f, 0.0f, nullptr);
    matmul256_kernel<<<256, 32, 0, stream>>>(T2, SigmaN, Pc, Pn, -0.5f, 1.5f, nullptr);
    float* t = Pc; Pc = Pn; Pn = t;
  }

  // M0 = sqrt(rTr) * (rot @ P)   (wm = P*sqrt(rTr), then rotation)
  matmul256_kernel<<<256, 32, 0, stream>>>(rot, Pc, nullptr, M0, 1.0f, 0.0f, scal + 1);

  fold_kernel<<<1, 256, 0, stream>>>(M0, mean, weight, bias, Mp, betav);

  // out = Mp @ x + betav: 4 tile-groups * 6272 m-chunks waves, 8 waves/block
  apply_kernel<<<(4 * (MTOT / 16)) / 8, 256, 0, stream>>>(X, Mp, betav, outp);
}